// UnfoldMatmulFold_28467043238111
// MI455X (gfx1250) — compile-verified
//
#include <hip/hip_runtime.h>

typedef __attribute__((ext_vector_type(2))) float v2f;
typedef __attribute__((ext_vector_type(8))) float v8f;
typedef __attribute__((__vector_size__(4 * sizeof(int)))) int i32x4;

#define KS   3
#define ST   2
#define PADN 1
#define HARD_H 28
#define HARD_W 28
#define NC   768
#define NH   12
#define DH   64
#define HH   14
#define WW   14
#define PP   9          // KS*KS
#define KP   12         // K padded to 3 WMMA steps of 4
#define MP   16         // M padded

#if defined(__has_builtin)
#if __has_builtin(__builtin_amdgcn_global_load_async_to_lds_b128)
#define HAVE_ASYNC_LDS 1
#else
#define HAVE_ASYNC_LDS 0
#endif
#if __has_builtin(__builtin_amdgcn_s_wait_asynccnt)
#define HAVE_WAIT_ASYNC 1
#else
#define HAVE_WAIT_ASYNC 0
#endif
#else
#define HAVE_ASYNC_LDS 0
#define HAVE_WAIT_ASYNC 0
#endif

#define AS1 __attribute__((address_space(1)))
#define AS3 __attribute__((address_space(3)))

static __device__ __forceinline__ void wait_async_zero() {
#if HAVE_ASYNC_LDS
#if HAVE_WAIT_ASYNC
    __builtin_amdgcn_s_wait_asynccnt(0);
#else
    asm volatile("s_wait_asynccnt 0" ::: "memory");
#endif
#endif
}

// ---------------------------------------------------------------------------
// Zero-init the output (harness poisons d_out with 0xAA).
// ---------------------------------------------------------------------------
__global__ void zero_out_kernel(float* __restrict__ out, int n4, int ntail) {
    int i = blockIdx.x * blockDim.x + threadIdx.x;
    if (i < n4) {
        float4 z = make_float4(0.f, 0.f, 0.f, 0.f);
        ((float4*)out)[i] = z;
    }
    if (i == 0) {
        for (int t = 0; t < ntail; ++t) out[(size_t)n4 * 4 + t] = 0.f;
    }
}

// ---------------------------------------------------------------------------
// One workgroup per (batch, patch) within a parity pass.
// 256 threads = 8 wave32.  48 WMMA tiles (12 heads x 4 chan-tiles), 6/wave.
// A and B are zero-padded in LDS so fragment gathers are unconditional.
// B rows stream in via GLOBAL_LOAD_ASYNC_TO_LDS_B128 (ASYNCcnt).
// C rows are folded back through LDS (overlaying the dead B rows) and
// RMW'd to global with coalesced float4 ops.  Patches within a parity pass
// have disjoint output pixels -> no atomics, deterministic accumulation.
// ---------------------------------------------------------------------------
__global__ __launch_bounds__(256) void patch_attn_fold_kernel(
    const float* __restrict__ vv,    // (B, 28, 28, 768)
    const float* __restrict__ attn,  // (B, 14, 14, 12, 9, 9)
    float*       __restrict__ out,   // (B, 28, 28, 768)
    int hp, int wp)                  // parity of h / w for this pass
{
    // v_lds: B matrix, 12 q-rows x 768 ch (rows 9..11 zero).  Reused later
    // as the 9 x 768 mm result staging area.                    36864 B
    __shared__ __align__(16) float v_lds[KP * NC];
    // a_pad: A matrix per head, 16 p-rows x 12 q-cols, zero padded. 9216 B
    __shared__ __align__(16) float a_pad[NH * MP * KP];

    const int b  = blockIdx.y;
    const int pi = blockIdx.x;               // 0..48  (7 x 7 patches per pass)
    const int h  = hp + 2 * (pi / 7);
    const int w  = wp + 2 * (pi % 7);
    const int tid = threadIdx.x;

    // ---- stage B: one row (3072 B) per iteration; y/x/bounds are uniform --
    #pragma unroll
    for (int q = 0; q < KP; ++q) {
        if (tid < NC / 4) {
            int y = ST * h + (q / KS) - PADN;   // block-uniform
            int x = ST * w + (q % KS) - PADN;   // block-uniform
            bool inb = (q < PP) && ((unsigned)y < HARD_H) && ((unsigned)x < HARD_W);
#if HAVE_ASYNC_LDS
            if (inb) {
                size_t g = (size_t)((b * HARD_H + y) * HARD_W + x) * NC + tid * 4;
                __builtin_amdgcn_global_load_async_to_lds_b128(
                    (AS1 i32x4*)(vv + g),
                    (AS3 i32x4*)(v_lds + q * NC + tid * 4), 0, 0);
            } else {
                ((float4*)v_lds)[q * (NC / 4) + tid] =
                    make_float4(0.f, 0.f, 0.f, 0.f);
            }
#else
            float4 val = make_float4(0.f, 0.f, 0.f, 0.f);
            if (inb) {
                size_t g = ((size_t)((b * HARD_H + y) * HARD_W + x) * NC) / 4 + tid;
                val = ((const float4*)vv)[g];
            }
            ((float4*)v_lds)[q * (NC / 4) + tid] = val;
#endif
        }
    }

    // ---- stage A: padded 16x12 per head; p/q math hoisted out of the loop -
    {
        size_t abase = (size_t)((b * HH + h) * WW + w) * (NH * PP * PP);
        int p = tid / KP;                     // 0..15 for tid < 192
        int q = tid - p * KP;                 // 0..11
        bool aval = (p < PP) && (q < PP);
        int aoff = p * PP + q;
        #pragma unroll
        for (int n = 0; n < NH; ++n) {
            if (tid < MP * KP) {
                float val = aval ? attn[abase + n * (PP * PP) + aoff] : 0.f;
                a_pad[n * (MP * KP) + tid] = val;
            }
        }
    }
    wait_async_zero();
    __syncthreads();

    const int wave = tid >> 5;
    const int lane = tid & 31;
    const int half = lane >> 4;   // 0: lanes 0-15, 1: lanes 16-31
    const int m    = lane & 15;   // M row / N column within half

    // ---- compute 6 tiles per wave; keep all accumulators live -------------
    v8f accs[6];
    #pragma unroll
    for (int ti = 0; ti < 6; ++ti) {
        int t  = wave * 6 + ti;
        int n  = t >> 2;                       // head 0..11
        int dt = t & 3;                        // 16-chan tile 0..3
        int cn = n * DH + dt * 16 + m;         // this lane's channel column
        const float* Arow = &a_pad[n * (MP * KP) + m * KP + 2 * half];
        const float* Bcol = &v_lds[2 * half * NC + cn];

        v8f acc = {};
        #pragma unroll
        for (int k = 0; k < 3; ++k) {          // q chunks of 4 (9 -> 12)
            // A fragment: A[m][q0], A[m][q0+1]  (adjacent -> ds_load_b64)
            v2f af = *(const v2f*)(Arow + 4 * k);
            // B fragment: B[q0][cn], B[q0+1][cn] (stride NC rows)
            v2f bf;
            bf.x = Bcol[(4 * k + 0) * NC];
            bf.y = Bcol[(4 * k + 1) * NC];
            acc = __builtin_amdgcn_wmma_f32_16x16x4_f32(
                false, af, false, bf, (short)0, acc, false, false);
        }
        accs[ti] = acc;
    }

    __syncthreads();   // all waves done reading B rows; reuse v_lds for mm

    // ---- dump valid C rows (p < 9) into LDS: mm[p][c] ---------------------
    #pragma unroll
    for (int ti = 0; ti < 6; ++ti) {
        int t  = wave * 6 + ti;
        int n  = t >> 2;
        int dt = t & 3;
        int cn = n * DH + dt * 16 + m;
        #pragma unroll
        for (int r = 0; r < 8; ++r) {
            int p = r + half * 8;
            if (p < PP) v_lds[p * NC + cn] = accs[ti][r];
        }
    }
    __syncthreads();

    // ---- cooperative fold: per-row uniform bounds, coalesced float4 RMW ---
    #pragma unroll
    for (int p = 0; p < PP; ++p) {
        int y = ST * h + (p / KS) - PADN;      // block-uniform
        int x = ST * w + (p % KS) - PADN;      // block-uniform
        if ((unsigned)y < HARD_H && (unsigned)x < HARD_W) {
            if (tid < NC / 4) {
                size_t g = ((size_t)((b * HARD_H + y) * HARD_W + x) * NC) / 4 + tid;
                float4 m4 = ((const float4*)v_lds)[p * (NC / 4) + tid];
                float4 o4 = ((float4*)out)[g];
                o4.x += m4.x; o4.y += m4.y; o4.z += m4.z; o4.w += m4.w;
                ((float4*)out)[g] = o4;        // unique writer within this pass
            }
        }
    }
}

extern "C" void kernel_launch(void* const* d_in, const int* in_sizes, int n_in,
                              void* d_out, int out_size, void* d_ws, size_t ws_size,
                              hipStream_t stream) {
    const float* vv   = (const float*)d_in[0];
    const float* attn = (const float*)d_in[1];
    float*       out  = (float*)d_out;

    const int B = in_sizes[0] / (HARD_H * HARD_W * NC);

    int n4    = out_size >> 2;
    int ntail = out_size & 3;
    zero_out_kernel<<<(n4 + 255) / 256, 256, 0, stream>>>(out, n4, ntail);

    dim3 grid(7 * 7, B);
    // 4 conflict-free parity passes; stream order makes fp accumulation
    // deterministic.
    for (int hp = 0; hp < 2; ++hp)
        for (int wp = 0; wp < 2; ++wp)
            patch_attn_fold_kernel<<<grid, 256, 0, stream>>>(vv, attn, out, hp, wp);
}